// GraphNN_56581899158174
// MI455X (gfx1250) — compile-verified
//
#include <hip/hip_runtime.h>
#include <hip/hip_bf16.h>

#define N_NODES 50000
#define N_EDGES 800000
#define N_GRAPHS 64
#define IN_CH 128
#define HID 256
#define OUT_CH 128

typedef float v2f __attribute__((ext_vector_type(2)));
typedef float v8f __attribute__((ext_vector_type(8)));

// ---------------------------------------------------------------------------
// Degree accumulation: deg[dst] += 1 (deg buffer pre-zeroed)
// ---------------------------------------------------------------------------
__global__ void deg_kernel(const long long* __restrict__ ei, float* __restrict__ deg) {
    int e = blockIdx.x * 256 + threadIdx.x;
    if (e < N_EDGES) atomicAdd(&deg[(int)ei[N_EDGES + e]], 1.0f);
}

// dis = rsqrt(deg + 1)   (in place)
__global__ void rsqrt_kernel(float* __restrict__ dis) {
    int i = blockIdx.x * 256 + threadIdx.x;
    if (i < N_NODES) dis[i] = rsqrtf(dis[i] + 1.0f);
}

// ---------------------------------------------------------------------------
// WMMA GEMM: out[N_NODES, 256] = A[N_NODES, K] @ W[K, 256],  K in {128, 256}
// block = 256 threads = 8 waves; block covers one 16-row M tile x all 256 cols
// (each wave computes two adjacent 16x16 N tiles, sharing its A fragment).
// Uses V_WMMA_F32_16X16X4_F32 (full f32 precision).
// ---------------------------------------------------------------------------
__global__ __launch_bounds__(256) void gemm_wmma(const float* __restrict__ A,
                                                 const float* __restrict__ W,
                                                 float* __restrict__ out, int K) {
    __shared__ float ldsA[16 * 260];          // padded stride (K+4) -> no bank conflicts
    const int stride = K + 4;
    const int m0  = blockIdx.x * 16;
    const int tid = threadIdx.x;

    // cooperative, coalesced load of the 16 x K A-tile
    for (int f = tid; f < 16 * K; f += 256) {
        int r = f / K, c = f - r * K;
        ldsA[r * stride + c] = A[(size_t)(m0 + r) * K + c];
    }
    __syncthreads();

    const int lane  = tid & 31;
    const int wave  = tid >> 5;               // 0..7
    const int n0    = wave * 32;              // two 16-wide N tiles: n0 and n0+16
    const int row   = lane & 15;              // M index for A, N index for B
    const int khalf = (lane >> 4) * 2;        // lanes 16-31 hold K=2,3 of each step

    v8f c0 = {};
    v8f c1 = {};
#pragma unroll 4
    for (int k0 = 0; k0 < K; k0 += 4) {
        const int kk = k0 + khalf;
        v2f a;
        a.x = ldsA[row * stride + kk];
        a.y = ldsA[row * stride + kk + 1];
        const float* wp = W + (size_t)kk * 256;
        v2f b0, b1;
        b0.x = wp[n0 + row];
        b0.y = wp[256 + n0 + row];
        b1.x = wp[n0 + 16 + row];
        b1.y = wp[256 + n0 + 16 + row];
        c0 = __builtin_amdgcn_wmma_f32_16x16x4_f32(false, a, false, b0, (short)0, c0, false, false);
        c1 = __builtin_amdgcn_wmma_f32_16x16x4_f32(false, a, false, b1, (short)0, c1, false, false);
    }

    // C/D layout: VGPR v -> M = v (lanes 0-15) / v+8 (lanes 16-31), N = lane&15
    const int nn = lane & 15;
    const int mb = m0 + ((lane >> 4) ? 8 : 0);
    float* o = out + (size_t)mb * 256 + n0 + nn;
#pragma unroll
    for (int v = 0; v < 8; ++v) {
        o[(size_t)v * 256]      = c0[v];
        o[(size_t)v * 256 + 16] = c1[v];
    }
}

// ---------------------------------------------------------------------------
// agg[i,c] = hW[i,c] * dis[i]^2 + b[c]   (self-loop + bias; fully initializes agg)
// ---------------------------------------------------------------------------
__global__ void self_init_kernel(const float* __restrict__ hW, const float* __restrict__ dis,
                                 const float* __restrict__ b, float* __restrict__ agg) {
    int t = blockIdx.x * 256 + threadIdx.x;   // over N_NODES * 64 (float4 per thread)
    int i = t >> 6;
    int c = (t & 63) << 2;
    float d = dis[i];
    float s = d * d;
    float4 hv = *(const float4*)(hW + (size_t)i * 256 + c);
    float4 bv = *(const float4*)(b + c);
    float4 o;
    o.x = hv.x * s + bv.x;
    o.y = hv.y * s + bv.y;
    o.z = hv.z * s + bv.z;
    o.w = hv.w * s + bv.w;
    *(float4*)(agg + (size_t)i * 256 + c) = o;
}

// ---------------------------------------------------------------------------
// Edge scatter: agg[dst] += hW[src] * dis[src]*dis[dst]  (L2-resident atomics)
// 64 threads per edge, float4 per thread.
// ---------------------------------------------------------------------------
__global__ void scatter_kernel(const long long* __restrict__ ei, const float* __restrict__ hW,
                               const float* __restrict__ dis, float* __restrict__ agg) {
    int t = blockIdx.x * 256 + threadIdx.x;   // over N_EDGES * 64
    int e = t >> 6;
    int c = (t & 63) << 2;
    int s = (int)ei[e];
    int d = (int)ei[N_EDGES + e];
    float norm = dis[s] * dis[d];
    float4 hv = *(const float4*)(hW + (size_t)s * 256 + c);
    float* ap = agg + (size_t)d * 256 + c;
    atomicAdd(ap + 0, hv.x * norm);
    atomicAdd(ap + 1, hv.y * norm);
    atomicAdd(ap + 2, hv.z * norm);
    atomicAdd(ap + 3, hv.w * norm);
}

__global__ void relu_kernel(float* __restrict__ h) {
    int t = blockIdx.x * 256 + threadIdx.x;   // over N_NODES * 64
    float4* p = (float4*)(h) + t;
    float4 v = *p;
    v.x = fmaxf(v.x, 0.f);
    v.y = fmaxf(v.y, 0.f);
    v.z = fmaxf(v.z, 0.f);
    v.w = fmaxf(v.w, 0.f);
    *p = v;
}

// ---------------------------------------------------------------------------
// Pooling (batch is sorted): run-length accumulate, flush one atomic per run.
// ---------------------------------------------------------------------------
__global__ void counts_kernel(const long long* __restrict__ batch, float* __restrict__ counts) {
    int i = blockIdx.x * 256 + threadIdx.x;
    if (i < N_NODES) atomicAdd(&counts[(int)batch[i]], 1.0f);
}

__global__ void pool_kernel(const float* __restrict__ h, const long long* __restrict__ batch,
                            float* __restrict__ pooled) {
    int c  = threadIdx.x;                 // channel 0..255
    int n0 = blockIdx.x * 128;
    int ne = n0 + 128;
    if (ne > N_NODES) ne = N_NODES;
    int cur = (int)batch[n0];
    float acc = 0.f;
    for (int n = n0; n < ne; ++n) {
        int g = (int)batch[n];
        if (g != cur) {
            atomicAdd(&pooled[(size_t)cur * 256 + c], acc);
            acc = 0.f;
            cur = g;
        }
        acc += h[(size_t)n * 256 + c];
    }
    atomicAdd(&pooled[(size_t)cur * 256 + c], acc);
}

__global__ void div_kernel(float* __restrict__ pooled, const float* __restrict__ counts) {
    int g = blockIdx.x, c = threadIdx.x;
    pooled[(size_t)g * 256 + c] /= fmaxf(counts[g], 1.0f);
}

// ---------------------------------------------------------------------------
// FC layers (tiny: 64 rows)
// ---------------------------------------------------------------------------
__global__ void fc1_kernel(const float* __restrict__ pooled, const float* __restrict__ hlr,
                           const float* __restrict__ stdv, const float* __restrict__ W,
                           const float* __restrict__ b, float* __restrict__ z) {
    int g = blockIdx.x, j = threadIdx.x;  // 64 x 256
    float acc = b[j];
    const float* pr = pooled + (size_t)g * 256;
    for (int k = 0; k < 256; ++k) acc += pr[k] * W[(size_t)k * 256 + j];
    acc += hlr[g] * W[(size_t)256 * 256 + j];
    acc += stdv[g] * W[(size_t)257 * 256 + j];
    z[(size_t)g * 256 + j] = fmaxf(acc, 0.f);
}

__global__ void fc2_kernel(const float* __restrict__ z, const float* __restrict__ W,
                           const float* __restrict__ b, float* __restrict__ out) {
    int g = blockIdx.x, j = threadIdx.x;  // 64 x 128
    float acc = b[j];
    const float* zr = z + (size_t)g * 256;
    for (int k = 0; k < 256; ++k) acc += zr[k] * W[(size_t)k * 128 + j];
    out[(size_t)g * 128 + j] = acc;
}

// ---------------------------------------------------------------------------
extern "C" void kernel_launch(void* const* d_in, const int* in_sizes, int n_in,
                              void* d_out, int out_size, void* d_ws, size_t ws_size,
                              hipStream_t stream) {
    const float*     x     = (const float*)d_in[0];
    const long long* ei    = (const long long*)d_in[1];
    const long long* batch = (const long long*)d_in[2];
    const float*     hlr   = (const float*)d_in[3];
    const float*     stdv  = (const float*)d_in[4];
    const float*     W0    = (const float*)d_in[5];
    const float*     b0    = (const float*)d_in[6];
    const float*     W1    = (const float*)d_in[7];
    const float*     b1    = (const float*)d_in[8];
    const float*     W2    = (const float*)d_in[9];
    const float*     b2    = (const float*)d_in[10];
    const float*     fcW0  = (const float*)d_in[11];
    const float*     fcb0  = (const float*)d_in[12];
    const float*     fcW1  = (const float*)d_in[13];
    const float*     fcb1  = (const float*)d_in[14];
    float*           out   = (float*)d_out;

    // workspace layout (floats)
    float* ws = (float*)d_ws;
    const size_t NF = (size_t)N_NODES * HID;      // 12.8M floats
    float* dis    = ws;               // 50176 reserved
    float* hW     = ws + 50176;
    float* aggA   = hW + NF;
    float* aggB   = aggA + NF;
    float* pooled = aggB + NF;        // 64*256
    float* counts = pooled + 16384;   // 64 (reserve 256)
    float* z      = counts + 256;     // 64*256

    const int B = 256;

    // degree / normalization
    hipMemsetAsync(dis, 0, (size_t)N_NODES * sizeof(float), stream);
    deg_kernel<<<(N_EDGES + B - 1) / B, B, 0, stream>>>(ei, dis);
    rsqrt_kernel<<<(N_NODES + B - 1) / B, B, 0, stream>>>(dis);

    const int gemm_blocks = N_NODES / 16;            // 3125 (exact)
    const int pw_blocks   = (N_NODES * 64) / B;      // 12500 (exact)
    const int sc_blocks   = (N_EDGES * 64) / B;      // 200000 (exact)

    // ---- layer 0: x[.,128] -> aggA ----
    gemm_wmma<<<gemm_blocks, B, 0, stream>>>(x, W0, hW, IN_CH);
    self_init_kernel<<<pw_blocks, B, 0, stream>>>(hW, dis, b0, aggA);
    scatter_kernel<<<sc_blocks, B, 0, stream>>>(ei, hW, dis, aggA);
    relu_kernel<<<pw_blocks, B, 0, stream>>>(aggA);

    // ---- layer 1: aggA -> aggB ----
    gemm_wmma<<<gemm_blocks, B, 0, stream>>>(aggA, W1, hW, HID);
    self_init_kernel<<<pw_blocks, B, 0, stream>>>(hW, dis, b1, aggB);
    scatter_kernel<<<sc_blocks, B, 0, stream>>>(ei, hW, dis, aggB);
    relu_kernel<<<pw_blocks, B, 0, stream>>>(aggB);

    // ---- layer 2: aggB -> aggA ----
    gemm_wmma<<<gemm_blocks, B, 0, stream>>>(aggB, W2, hW, HID);
    self_init_kernel<<<pw_blocks, B, 0, stream>>>(hW, dis, b2, aggA);
    scatter_kernel<<<sc_blocks, B, 0, stream>>>(ei, hW, dis, aggA);
    relu_kernel<<<pw_blocks, B, 0, stream>>>(aggA);

    // ---- pooling ----
    hipMemsetAsync(pooled, 0, 16384 * sizeof(float), stream);
    hipMemsetAsync(counts, 0, 64 * sizeof(float), stream);
    counts_kernel<<<(N_NODES + B - 1) / B, B, 0, stream>>>(batch, counts);
    pool_kernel<<<(N_NODES + 127) / 128, B, 0, stream>>>(aggA, batch, pooled);
    div_kernel<<<N_GRAPHS, 256, 0, stream>>>(pooled, counts);

    // ---- MLP head ----
    fc1_kernel<<<N_GRAPHS, 256, 0, stream>>>(pooled, hlr, stdv, fcW0, fcb0, z);
    fc2_kernel<<<N_GRAPHS, 128, 0, stream>>>(z, fcW1, fcb1, out);
}